// DWT_55490977464395
// MI455X (gfx1250) — compile-verified
//
#include <hip/hip_runtime.h>

// 2-D Haar DWT, single level, fp32.
// Input : x  (8,64,512,512)  = 512 images of 512x512
// Output: (ll,lh,hl,hh) each (8,64,256,256), concatenated flat in d_out.
//
// Bandwidth-bound: 1 GB total traffic -> ~46us floor @ 23.3 TB/s. WMMA is a
// pessimization here (AI ~0.1 FLOP/byte). gfx1250 paths: non-temporal
// load/store hints (one-touch 1GB stream must not churn the 192MB L2/WGP$)
// and global_prefetch_b8 at a fixed stream distance.
//
// Work decomposition (fixed shape): 2^21 threads; thread -> (img0, r, c):
//   c    = tid & 63      (8 input cols / 4 output cols per thread)
//   r    = (tid>>6)&255  (row pair)
//   img0 = tid>>14       (0..127); thread handles images img0 + {0,128,256,384}
// Stride between iterations is a compile-time constant -> loop fully
// strength-reduced to pointer adds (minimal valu/salu per 128B moved).

typedef float v4f __attribute__((ext_vector_type(4)));

__global__ __launch_bounds__(256) void
haar2d_dwt_kernel(const float* __restrict__ x, float* __restrict__ out)
{
    const float s = 0.70710678118654752440f;     // reference INV_SQRT2

    constexpr long long PLANE = 512LL * 256 * 256;       // elems per subband
    constexpr long long ISTEP = 128LL * 512 * 512;       // input elems / iter
    constexpr long long OSTEP = 128LL * 256 * 256;       // output elems / iter

    float* __restrict__ ll = out;
    float* __restrict__ lh = out + PLANE;
    float* __restrict__ hl = out + 2 * PLANE;
    float* __restrict__ hh = out + 3 * PLANE;

    const int tid  = blockIdx.x * 256 + threadIdx.x;     // 0 .. 2^21-1
    const int c    = tid & 63;
    const int r    = (tid >> 6) & 255;
    const int img0 = tid >> 14;

    long long ibase = (((long long)img0 * 512) + 2 * r) * 512 + (long long)c * 8;
    long long obase = (((long long)img0 * 256) + r) * 256 + (long long)c * 4;

#pragma unroll
    for (int k = 0; k < 4; ++k) {
        if (k < 3) {
            // Constant-distance stream prefetch of next iteration's two rows.
            __builtin_prefetch(x + ibase + ISTEP, 0, 1);
            __builtin_prefetch(x + ibase + ISTEP + 512, 0, 1);
        }

        // 128-bit non-temporal loads: 8 floats from row 2r and row 2r+1.
        v4f r0a = __builtin_nontemporal_load((const v4f*)(x + ibase));
        v4f r0b = __builtin_nontemporal_load((const v4f*)(x + ibase + 4));
        v4f r1a = __builtin_nontemporal_load((const v4f*)(x + ibase + 512));
        v4f r1b = __builtin_nontemporal_load((const v4f*)(x + ibase + 512 + 4));

        // De-interleave even/odd columns.
        v4f a  = {r0a.x, r0a.z, r0b.x, r0b.z};   // row0 even
        v4f b  = {r0a.y, r0a.w, r0b.y, r0b.w};   // row0 odd
        v4f cc = {r1a.x, r1a.z, r1b.x, r1b.z};   // row1 even
        v4f d  = {r1a.y, r1a.w, r1b.y, r1b.w};   // row1 odd

        // Row pass then column pass, each x(1/sqrt2): matches reference.
        v4f rl0 = (a + b)  * s;
        v4f rh0 = (a - b)  * s;
        v4f rl1 = (cc + d) * s;
        v4f rh1 = (cc - d) * s;

        v4f vll = (rl0 + rl1) * s;
        v4f vhl = (rl0 - rl1) * s;
        v4f vlh = (rh0 + rh1) * s;
        v4f vhh = (rh0 - rh1) * s;

        // 128-bit non-temporal stores to the four subband planes.
        __builtin_nontemporal_store(vll, (v4f*)(ll + obase));
        __builtin_nontemporal_store(vlh, (v4f*)(lh + obase));
        __builtin_nontemporal_store(vhl, (v4f*)(hl + obase));
        __builtin_nontemporal_store(vhh, (v4f*)(hh + obase));

        ibase += ISTEP;
        obase += OSTEP;
    }
}

extern "C" void kernel_launch(void* const* d_in, const int* in_sizes, int n_in,
                              void* d_out, int out_size, void* d_ws, size_t ws_size,
                              hipStream_t stream) {
    const float* x = (const float*)d_in[0];
    float* out = (float*)d_out;

    // Fixed reference shape: 134,217,728 input elems -> 2^21 threads x 4 iters.
    const int block = 256;   // 8 wave32 waves per workgroup
    const int grid  = 8192;  // 2,097,152 threads

    haar2d_dwt_kernel<<<grid, block, 0, stream>>>(x, out);
}